// my_loss_func_46462956208596
// MI455X (gfx1250) — compile-verified
//
#include <hip/hip_runtime.h>
#include <math.h>

// Problem constants (from reference)
#define BB 131072
#define CC 1000
#define CHUNKS 250           // CC / 4 float4 chunks per row (row = 4000B, 16B aligned)
#define ROWS_PER_BLOCK 8     // one wave32 per row
#define ALPH 1.0f
#define TAO  1.0f

typedef float    f4    __attribute__((ext_vector_type(4)));
typedef unsigned u32x4 __attribute__((ext_vector_type(4)));
typedef unsigned u32x8 __attribute__((ext_vector_type(8)));

// ---- CDNA5 Tensor Data Mover: 1-D tile (4000B row) global -> LDS ----
// D# built per CDNA5 ISA Sec 8.3/8.4; groups 2/3 omitted (<=2D tensor).
__device__ __forceinline__ void tdm_load_row_to_lds(unsigned lds_addr,
                                                    const float* gptr) {
  unsigned long long ga = (unsigned long long)gptr;
  u32x4 g0;
  g0[0] = 1u;                                   // count=1, user descriptor
  g0[1] = lds_addr;                             // lds_addr (bytes)
  g0[2] = (unsigned)(ga & 0xffffffffull);       // global_addr[31:0]
  g0[3] = (unsigned)((ga >> 32) & 0x1ffffffu)   // global_addr[56:32]
        | (2u << 30);                           // type=2 ("image")
  u32x8 g1;
  g1[0] = 2u << 16;                             // wg_mask=0, data_size=2 (4B), no pad/iter
  g1[1] = (CC & 0xffffu) << 16;                 // tensor_dim0[15:0]      (=1000)
  g1[2] = (CC >> 16) | (1u << 16);              // tensor_dim0[31:16], tensor_dim1=1
  g1[3] = (CC & 0xffffu) << 16;                 // tensor_dim1 hi=0, tile_dim0=1000
  g1[4] = 0u;                                   // tile_dim1=0 (unused), tile_dim2=0
  g1[5] = CC;                                   // tensor_dim0_stride[31:0] (=1000)
  g1[6] = 0u;                                   // stride hi, tensor_dim1_stride lo
  g1[7] = 0u;                                   // tensor_dim1_stride hi
  asm volatile("tensor_load_to_lds %0, %1" :: "s"(g0), "s"(g1) : "memory");
}
__device__ __forceinline__ void wait_tensor0() {
  asm volatile("s_wait_tensorcnt 0x0" ::: "memory");
}

__global__ __launch_bounds__(256)
void row_loss_kernel(const float* __restrict__ score,
                     const float* __restrict__ label,
                     float* __restrict__ block_partials) {
  // 4KB LDS staging per wave (1000 floats used) + per-wave loss slots
  __shared__ __align__(16) float s_score[ROWS_PER_BLOCK][1024];
  __shared__ float s_loss[ROWS_PER_BLOCK];

  const int lane = threadIdx.x & 31;
  const int wv   = threadIdx.x >> 5;
  const int row  = blockIdx.x * ROWS_PER_BLOCK + wv;
  const size_t base = (size_t)row * CC;

  // ---- 1) kick off TDM DMA of this wave's score row into LDS (one instruction)
  {
    unsigned lds_addr = (unsigned)(unsigned long long)(&s_score[wv][0]);
    tdm_load_row_to_lds(lds_addr, score + base);
  }

  // ---- 2) label argmax (global, non-temporal), overlapped with the TDM transfer
  const f4* lrow = (const f4*)(label + base);
  float bestv = -__builtin_inff();
  int   besti = 0;
  for (int j = lane; j < CHUNKS; j += 32) {
    f4 v = __builtin_nontemporal_load(&lrow[j]);
    int c = j * 4;
    if (v.x > bestv) { bestv = v.x; besti = c;     }
    if (v.y > bestv) { bestv = v.y; besti = c + 1; }
    if (v.z > bestv) { bestv = v.z; besti = c + 2; }
    if (v.w > bestv) { bestv = v.w; besti = c + 3; }
  }
  #pragma unroll
  for (int m = 16; m > 0; m >>= 1) {       // wave32 argmax reduce, first-occurrence tiebreak
    float ov = __shfl_xor(bestv, m, 32);
    int   oi = __shfl_xor(besti, m, 32);
    if (ov > bestv || (ov == bestv && oi < besti)) { bestv = ov; besti = oi; }
  }
  const int idx = besti;                   // uniform across wave

  // ---- 3) wait for the tile, grab s_y, then mask the true class in LDS.
  // Same-wave LDS ops are in order (DScnt) -> no barrier needed.
  wait_tensor0();
  const float s_y = s_score[wv][idx];
  if (lane == 0) s_score[wv][idx] = -__builtin_inff();

  // ---- 4) pass 1: row max (true class already -inf)
  const f4* srow_lds = (const f4*)(&s_score[wv][0]);
  float M = -__builtin_inff();
  for (int j = lane; j < CHUNKS; j += 32) {
    f4 v = srow_lds[j];
    M = fmaxf(M, fmaxf(fmaxf(v.x, v.y), fmaxf(v.z, v.w)));
  }
  #pragma unroll
  for (int m = 16; m > 0; m >>= 1) M = fmaxf(M, __shfl_xor(M, m, 32));

  // ---- 5) pass 2: sum exp(v - M); true class contributes exp(-inf)=0
  float S = 0.f;
  for (int j = lane; j < CHUNKS; j += 32) {
    f4 v = srow_lds[j];
    S += __expf(v.x - M) + __expf(v.y - M) + __expf(v.z - M) + __expf(v.w - M);
  }
  #pragma unroll
  for (int m = 16; m > 0; m >>= 1) S += __shfl_xor(S, m, 32);

  // loss = TAO * logsumexp((ALPH + s_i - s_y)/TAO) ; with TAO==1:
  if (lane == 0) s_loss[wv] = (ALPH - s_y) + M + logf(S);

  __syncthreads();
  if (threadIdx.x == 0) {
    float t = 0.f;
    #pragma unroll
    for (int i = 0; i < ROWS_PER_BLOCK; ++i) t += s_loss[i];
    block_partials[blockIdx.x] = t;   // deterministic per-block partial
  }
}

__global__ __launch_bounds__(1024)
void reduce_partials_kernel(const float* __restrict__ partials, int n,
                            float* __restrict__ out, float inv_b) {
  __shared__ float sm[1024];
  const int tid = threadIdx.x;
  float s = 0.f;
  for (int i = tid; i < n; i += 1024) s += partials[i];
  sm[tid] = s;
  __syncthreads();
  #pragma unroll
  for (int w = 512; w > 0; w >>= 1) {
    if (tid < w) sm[tid] += sm[tid + w];
    __syncthreads();
  }
  if (tid == 0) out[0] = sm[0] * inv_b;
}

extern "C" void kernel_launch(void* const* d_in, const int* in_sizes, int n_in,
                              void* d_out, int out_size, void* d_ws, size_t ws_size,
                              hipStream_t stream) {
  const float* score = (const float*)d_in[0];
  const float* label = (const float*)d_in[1];
  float* out      = (float*)d_out;
  float* partials = (float*)d_ws;          // 16384 floats = 64 KB scratch

  const int nblk = BB / ROWS_PER_BLOCK;    // 16384 blocks, 8 waves each
  row_loss_kernel<<<nblk, 256, 0, stream>>>(score, label, partials);
  reduce_partials_kernel<<<1, 1024, 0, stream>>>(partials, nblk, out, 1.0f / (float)BB);
}